// TripletMarginLoss_81767587381279
// MI455X (gfx1250) — compile-verified
//
#include <hip/hip_runtime.h>
#include <hip/hip_bf16.h>

typedef __attribute__((ext_vector_type(2))) float v2f;
typedef __attribute__((ext_vector_type(8))) float v8f;

#define B_SAMPLES 131072
#define DIM 256
#define NCLS 100
#define WAVES_PER_BLOCK 4
#define SAMPLES_PER_WAVE 16
#define CHUNK 64                 // dims processed per chunk
#define STEPS (CHUNK / 4)        // WMMA K=4 steps per chunk
#define LDS_STRIDE 33            // padded float2 stride per K-step (bank-conflict free)

// ---------------------------------------------------------------------------
// Main kernel: one wave computes d_ap^2 / d_an^2 for 16 samples using
// V_WMMA_F32_16X16X4_F32 with an all-ones B matrix (row-sum reduction),
// then finishes the hinge loss and atomically accumulates (loss_sum, cnt).
// ---------------------------------------------------------------------------
__global__ __launch_bounds__(128) void triplet_main_kernel(
    const float* __restrict__ emb,      // [B, D]
    const float* __restrict__ margin,   // [C, C]
    const int*   __restrict__ labels,   // [B]
    const int*   __restrict__ pos_idx,  // [B]
    const int*   __restrict__ neg_idx,  // [B]
    const int*   __restrict__ valid,    // [B]
    float*       __restrict__ ws)       // ws[0]=loss sum, ws[1]=valid count
{
    __shared__ v2f   lds_ap[WAVES_PER_BLOCK][STEPS * LDS_STRIDE];
    __shared__ v2f   lds_an[WAVES_PER_BLOCK][STEPS * LDS_STRIDE];
    __shared__ float lds_dap[WAVES_PER_BLOCK][16];
    __shared__ float lds_dan[WAVES_PER_BLOCK][16];

    const int tid  = threadIdx.x;
    const int wave = tid >> 5;
    const int lane = tid & 31;
    const int groupBase = blockIdx.x * (WAVES_PER_BLOCK * SAMPLES_PER_WAVE)
                        + wave * SAMPLES_PER_WAVE;

    v8f cap = {};   // accumulates sum((a-p)^2) per sample (16x16 C, col-replicated)
    v8f can = {};   // accumulates sum((a-n)^2)

    const v2f ones = {1.0f, 1.0f};

    // A-matrix staging position for this lane: K-pair index within chunk.
    // lane l holds dims {2l, 2l+1} of the current 64-dim chunk:
    //   K-step  s = l >> 1,  pair-select = l & 1  ->  A lane = m + 16*(l&1)
    const int sl  = lane >> 1;
    const int psel = lane & 1;

    for (int chunk = 0; chunk < DIM / CHUNK; ++chunk) {
        const int kbase = chunk * CHUNK + 2 * lane;

        // ---- phase 1: coalesced loads + diff^2, staged in WMMA A layout ----
        for (int m = 0; m < SAMPLES_PER_WAVE; ++m) {
            const int i  = groupBase + m;
            const int pi = pos_idx[i];
            const int ni = neg_idx[i];
            v2f a = *(const v2f*)(emb + (size_t)i  * DIM + kbase);
            v2f p = *(const v2f*)(emb + (size_t)pi * DIM + kbase);
            v2f n = *(const v2f*)(emb + (size_t)ni * DIM + kbase);
            v2f dap = a - p; dap *= dap;
            v2f dan = a - n; dan *= dan;
            const int dst = sl * LDS_STRIDE + m + 16 * psel;
            lds_ap[wave][dst] = dap;
            lds_an[wave][dst] = dan;
        }
        __syncthreads();

        // ---- phase 2: WMMA row-sum accumulation over this chunk ----
        #pragma unroll
        for (int s = 0; s < STEPS; ++s) {
            v2f aAp = lds_ap[wave][s * LDS_STRIDE + lane];
            v2f aAn = lds_an[wave][s * LDS_STRIDE + lane];
            cap = __builtin_amdgcn_wmma_f32_16x16x4_f32(
                      false, aAp, false, ones, (short)0, cap, false, false);
            can = __builtin_amdgcn_wmma_f32_16x16x4_f32(
                      false, aAn, false, ones, (short)0, can, false, false);
        }
        __syncthreads();
    }

    // ---- extract column N=0 of the C matrices ----
    // C layout: lane l, VGPR v  ->  M = v + 8*(l>=16), N = l&15
    if (lane == 0) {
        #pragma unroll
        for (int v = 0; v < 8; ++v) {
            lds_dap[wave][v] = cap[v];
            lds_dan[wave][v] = can[v];
        }
    }
    if (lane == 16) {
        #pragma unroll
        for (int v = 0; v < 8; ++v) {
            lds_dap[wave][8 + v] = cap[v];
            lds_dan[wave][8 + v] = can[v];
        }
    }
    __syncthreads();

    // ---- hinge loss for lanes 0..15, wave reduction, atomic accumulate ----
    float loss = 0.0f, cnt = 0.0f;
    if (lane < 16) {
        const int i   = groupBase + lane;
        const float dap = sqrtf(lds_dap[wave][lane]);
        const float dan = sqrtf(lds_dan[wave][lane]);
        const int ni  = neg_idx[i];
        const float mg = margin[labels[i] * NCLS + labels[ni]];
        const float w  = valid[i] ? 1.0f : 0.0f;
        loss = fmaxf(dap - dan + mg, 0.0f) * w;
        cnt  = w;
    }
    #pragma unroll
    for (int off = 16; off > 0; off >>= 1) {
        loss += __shfl_xor(loss, off, 32);
        cnt  += __shfl_xor(cnt,  off, 32);
    }
    if (lane == 0) {
        atomicAdd(&ws[0], loss);
        atomicAdd(&ws[1], cnt);
    }
}

__global__ void triplet_init_kernel(float* __restrict__ ws) {
    ws[0] = 0.0f;
    ws[1] = 0.0f;
}

__global__ void triplet_finalize_kernel(const float* __restrict__ ws,
                                        float* __restrict__ out) {
    out[0] = ws[0] / fmaxf(ws[1], 1.0f);
}

extern "C" void kernel_launch(void* const* d_in, const int* in_sizes, int n_in,
                              void* d_out, int out_size, void* d_ws, size_t ws_size,
                              hipStream_t stream) {
    const float* emb     = (const float*)d_in[0];
    const float* margin  = (const float*)d_in[1];
    const int*   labels  = (const int*)d_in[2];
    const int*   pos_idx = (const int*)d_in[3];
    const int*   neg_idx = (const int*)d_in[4];
    const int*   valid   = (const int*)d_in[5];
    float* out = (float*)d_out;
    float* ws  = (float*)d_ws;

    (void)in_sizes; (void)n_in; (void)out_size; (void)ws_size;

    triplet_init_kernel<<<1, 1, 0, stream>>>(ws);

    const int samples_per_block = WAVES_PER_BLOCK * SAMPLES_PER_WAVE;   // 64
    const int grid = B_SAMPLES / samples_per_block;                     // 2048
    triplet_main_kernel<<<grid, 128, 0, stream>>>(
        emb, margin, labels, pos_idx, neg_idx, valid, ws);

    triplet_finalize_kernel<<<1, 1, 0, stream>>>(ws, out);
}